// ChamferLoss_7155415515637
// MI455X (gfx1250) — compile-verified
//
#include <hip/hip_runtime.h>
#include <hip/hip_bf16.h>

typedef __attribute__((ext_vector_type(2))) float v2f;
typedef __attribute__((ext_vector_type(8))) float v8f;

// Zero the 8 per-batch accumulators (harness poisons d_out before timing).
__global__ void chamfer_zero_kernel(float* __restrict__ out, int n) {
    int i = threadIdx.x;
    if (i < n) out[i] = 0.0f;
}

// One-sided Chamfer: for each point x_i in X[b], min_j ||x_i - Y[b,j]||^2,
// summed over i, atomically accumulated into out[b].
// Each wave (32 lanes, wave32) owns a 16-row X tile and sweeps Y in 16-point
// chunks; the 16x16 cross-dot tile comes from v_wmma_f32_16x16x4_f32 (K=3
// padded to 4 with zeros).
__global__ __launch_bounds__(256) void chamfer_one_sided_kernel(
    const float* __restrict__ X,   // [B, N, 3]  (rows: the "min per row" set)
    const float* __restrict__ Y,   // [B, M, 3]
    float* __restrict__ out,       // [B]
    int N, int M)
{
    const int b    = blockIdx.y;
    const int lane = threadIdx.x & 31;
    const int wave = threadIdx.x >> 5;
    const int i0   = (blockIdx.x * 8 + wave) * 16;   // 8 waves * 16 rows / block
    const int half = lane >> 4;                      // 0: K={0,1}, 1: K={2,3}
    const int n    = lane & 15;                      // row/col index within tile

    const float* Xb = X + (size_t)b * N * 3;
    const float* Yb = Y + (size_t)b * M * 3;

    // ---- A tile (16x4 f32): lane L holds row M = L%16.
    // half 0: VGPR{0,1} = (K0,K1) = (x,y); half 1: (K2,K3) = (z, 0-pad).
    const float ax = Xb[(size_t)(i0 + n) * 3 + 0];
    const float ay = Xb[(size_t)(i0 + n) * 3 + 1];
    const float az = Xb[(size_t)(i0 + n) * 3 + 2];
    const float rx = ax * ax + ay * ay + az * az;

    v2f A;
    A.x = half ? az : ax;
    A.y = half ? 0.0f : ay;

    // D-matrix slot v on lane L corresponds to row M = v + 8*half.
    // Gather rx for that row (lane r and r+16 both hold rx of tile row r).
    float rxl[8];
#pragma unroll
    for (int v = 0; v < 8; ++v)
        rxl[v] = __shfl(rx, v + (half << 3), 32);

    float runmin[8];
#pragma unroll
    for (int v = 0; v < 8; ++v) runmin[v] = 3.402823466e38f;

    for (int j0 = 0; j0 < M; j0 += 16) {
        // ---- B tile (4x16 f32): lane L holds column N = L%16, same K split.
        const float bx = Yb[(size_t)(j0 + n) * 3 + 0];
        const float by = Yb[(size_t)(j0 + n) * 3 + 1];
        const float bz = Yb[(size_t)(j0 + n) * 3 + 2];
        // Prefetch next chunk (speculative; dropped if past end of buffer).
        __builtin_prefetch(&Yb[(size_t)(j0 + 16 + n) * 3], 0, 0);

        const float ry = bx * bx + by * by + bz * bz;
        v2f Bm;
        Bm.x = half ? bz : bx;
        Bm.y = half ? 0.0f : by;

        // zz[v] = dot(X[i0 + v + 8*half], Y[j0 + lane%16])  via WMMA, C = 0.
        v8f zz = {};
        zz = __builtin_amdgcn_wmma_f32_16x16x4_f32(
            /*neg_a=*/false, A, /*neg_b=*/false, Bm,
            /*c_mod=*/(short)0, zz, /*reuse_a=*/false, /*reuse_b=*/false);

        // P = rx + ry - 2*zz ; track running min per D slot.
#pragma unroll
        for (int v = 0; v < 8; ++v) {
            const float p = fmaf(zz[v], -2.0f, rxl[v] + ry);
            runmin[v] = fminf(runmin[v], p);
        }
    }

    // Row-min: reduce each slot across the 16 lanes of its half.
#pragma unroll
    for (int v = 0; v < 8; ++v) {
        float m = runmin[v];
        m = fminf(m, __shfl_xor(m, 1, 32));
        m = fminf(m, __shfl_xor(m, 2, 32));
        m = fminf(m, __shfl_xor(m, 4, 32));
        m = fminf(m, __shfl_xor(m, 8, 32));
        runmin[v] = m;
    }
    // Sum the 8 row-mins this half owns, then add the other half's sum.
    float s = 0.0f;
#pragma unroll
    for (int v = 0; v < 8; ++v) s += runmin[v];
    s += __shfl_xor(s, 16, 32);

    if (lane == 0) atomicAdd(&out[b], s);
}

extern "C" void kernel_launch(void* const* d_in, const int* in_sizes, int n_in,
                              void* d_out, int out_size, void* d_ws, size_t ws_size,
                              hipStream_t stream) {
    const float* preds = (const float*)d_in[0];   // [8, 4096, 3]
    const float* gts   = (const float*)d_in[1];   // [8, 4096, 3]
    float* out = (float*)d_out;                   // [8]

    const int B = 8;
    const int N = in_sizes[0] / (B * 3);          // 4096
    (void)n_in; (void)d_ws; (void)ws_size;

    chamfer_zero_kernel<<<1, 32, 0, stream>>>(out, out_size);

    dim3 grid(N / 128, B);                        // 8 waves * 16 rows = 128 rows/block
    // loss_2: rows = gts, min over preds
    chamfer_one_sided_kernel<<<grid, 256, 0, stream>>>(gts, preds, out, N, N);
    // loss_1: rows = preds, min over gts
    chamfer_one_sided_kernel<<<grid, 256, 0, stream>>>(preds, gts, out, N, N);
}